// SpatialAttention_66391604462131
// MI455X (gfx1250) — compile-verified
//
#include <hip/hip_runtime.h>
#include <hip/hip_bf16.h>
#include <math.h>

// Problem dims (fixed by setup_inputs)
#define B_   32
#define C_   256
#define H_   56
#define W_   56
#define HW_  (H_ * W_)        // 3136
#define NPIX (B_ * HW_)       // 100352

typedef __attribute__((ext_vector_type(2))) float v2f;
typedef __attribute__((ext_vector_type(4))) float v4f;
typedef __attribute__((ext_vector_type(8))) float v8f;

// ---------------------------------------------------------------------------
// Kernel 1: channel-wise mean + max.
// Block = 256 threads = 64 pixels x 4 channel-quarters; LDS combine.
// Loads are fully coalesced (consecutive lanes -> consecutive hw) and use the
// default RT temporal hint so x becomes L2-resident (102.8 MB < 192 MB L2)
// for reuse by the multiply kernel.
// ---------------------------------------------------------------------------
__global__ __launch_bounds__(256) void reduce_kernel(const float* __restrict__ x,
                                                     float2* __restrict__ y) {
  __shared__ float ssum[256];
  __shared__ float smax[256];
  const int t  = threadIdx.x;
  const int p  = t & 63;        // pixel within block
  const int q  = t >> 6;        // channel quarter 0..3
  const int gp = blockIdx.x * 64 + p;           // global pixel (never spans batch: 3136 % 64 == 0)
  const int b  = gp / HW_;
  const int hw = gp - b * HW_;

  const float* xp = x + (size_t)(b * C_ + q * 64) * HW_ + hw;
  float s  = 0.0f;
  float mx = -3.402823466e38f;
#pragma unroll 8
  for (int c = 0; c < 64; ++c) {
    float v = xp[(size_t)c * HW_];
    s += v;
    mx = fmaxf(mx, v);
  }
  ssum[t] = s;
  smax[t] = mx;
  __syncthreads();
  if (q == 0) {
    float s0 = ssum[p] + ssum[p + 64] + ssum[p + 128] + ssum[p + 192];
    float m0 = fmaxf(fmaxf(smax[p], smax[p + 64]), fmaxf(smax[p + 128], smax[p + 192]));
    float2 r;
    r.x = s0 * (1.0f / (float)C_);  // mean
    r.y = m0;                       // max
    y[gp] = r;
  }
}

// ---------------------------------------------------------------------------
// Kernel 2: 7x7 conv over the 2-channel (avg,max) map + sigmoid, via
// V_WMMA_F32_16X16X4_F32. One wave per 16 output pixels (M=16).
// K = 2*tap + ch, tap in [0,49), ch in {0=avg,1=max}; K padded 98 -> 100
// (25 WMMA chunks of K=4). B broadcasts the weights across all N columns,
// so every column of D holds the same 16 conv outputs.
//
// A layout (32-bit 16x4): lanes 0-15 hold M=lane, K = kb+{0,1};
//                         lanes 16-31 hold M=lane-16, K = kb+{2,3}.
// => each lane's two A VGPRs are exactly one tap's (avg,max) pair.
// B layout (32-bit 4x16): symmetric, V0/V1 = K = (lane<16 ? kb : kb+2)+{0,1}.
// D layout: VGPR r: lanes 0-15 -> M=r (N=lane), lanes 16-31 -> M=8+r.
// ---------------------------------------------------------------------------
__global__ __launch_bounds__(32) void conv_wmma_kernel(const float2* __restrict__ y,
                                                       const float* __restrict__ cw, // [2*49]
                                                       float* __restrict__ att) {
  const int lane = threadIdx.x;       // 0..31
  const int m    = lane & 15;         // pixel row of A held by this lane
  const int hi   = lane >> 4;         // which K-half of the chunk

  const int gp = blockIdx.x * 16 + m;           // NPIX % 16 == 0, no remainder
  const int b  = gp / HW_;
  const int hw = gp - b * HW_;
  const int h  = hw / W_;
  const int w  = hw - h * W_;
  const float2* yb = y + (size_t)b * HW_;

  v8f acc = {0.f, 0.f, 0.f, 0.f, 0.f, 0.f, 0.f, 0.f};

  for (int c = 0; c < 25; ++c) {
    const int tap = 2 * c + hi;       // tap handled by this lane (A and B agree)
    float2 av = make_float2(0.f, 0.f);
    float  w0 = 0.f, w1 = 0.f;
    if (tap < 49) {
      const int kh = tap / 7;
      const int kw = tap - kh * 7;
      const int ih = h + kh - 3;
      const int iw = w + kw - 3;
      if (ih >= 0 && ih < H_ && iw >= 0 && iw < W_) av = yb[ih * W_ + iw];
      w0 = cw[tap];        // channel 0 (avg) weight
      w1 = cw[49 + tap];   // channel 1 (max) weight
    }
    v2f A = {av.x, av.y};
    v2f Bm = {w0, w1};
    acc = __builtin_amdgcn_wmma_f32_16x16x4_f32(
        /*neg_a=*/false, A, /*neg_b=*/false, Bm,
        /*c_mod=*/(short)0, acc, /*reuse_a=*/false, /*reuse_b=*/false);
  }

  // All N columns identical; column 0 lives in lanes 0 (M=0..7) and 16 (M=8..15).
  if (m == 0) {
    const int mbase = hi * 8;
#pragma unroll
    for (int r = 0; r < 8; ++r) {
      float z = acc[r];
      float s = 1.0f / (1.0f + expf(-z));
      att[blockIdx.x * 16 + mbase + r] = s;
    }
  }
}

// ---------------------------------------------------------------------------
// Kernel 3: out = x * att (broadcast over channels), b128 streaming.
// x reads are non-temporal (last-use: hit the L2-resident copy from kernel 1
// without promoting); out stores are non-temporal (never re-read: don't let
// the 102.8 MB store stream evict x from L2 mid-kernel).
// ---------------------------------------------------------------------------
__global__ __launch_bounds__(256) void mul_kernel(const float* __restrict__ x,
                                                  const float* __restrict__ att,
                                                  float* __restrict__ out, int n4) {
  const int i = blockIdx.x * blockDim.x + threadIdx.x;
  if (i >= n4) return;
  const int per_b = C_ * (HW_ / 4);             // float4 groups per batch image
  const int b     = i / per_b;
  const int rem   = i - b * per_b;
  const int hw4   = rem % (HW_ / 4);            // float4 group within the HxW plane

  const v4f xv = __builtin_nontemporal_load(reinterpret_cast<const v4f*>(x) + i);
  const v4f av = *(reinterpret_cast<const v4f*>(att + (size_t)b * HW_) + hw4);
  v4f o = xv * av;
  __builtin_nontemporal_store(o, reinterpret_cast<v4f*>(out) + i);
}

// ---------------------------------------------------------------------------
extern "C" void kernel_launch(void* const* d_in, const int* in_sizes, int n_in,
                              void* d_out, int out_size, void* d_ws, size_t ws_size,
                              hipStream_t stream) {
  const float* x  = (const float*)d_in[0];   // [32,256,56,56]
  const float* cw = (const float*)d_in[1];   // [1,2,7,7] -> 98 floats
  float* out      = (float*)d_out;

  // Workspace: (avg,max) pairs then attention map.
  float2* y   = (float2*)d_ws;                                   // NPIX float2 (802,816 B)
  float*  att = (float*)((char*)d_ws + (size_t)NPIX * sizeof(float2)); // NPIX floats (401,408 B)

  (void)in_sizes; (void)n_in; (void)out_size; (void)ws_size;

  // 1) channel mean+max reduce (fills L2 with x)
  reduce_kernel<<<NPIX / 64, 256, 0, stream>>>(x, y);

  // 2) 7x7 conv + sigmoid via WMMA (one wave per 16 output pixels)
  conv_wmma_kernel<<<NPIX / 16, 32, 0, stream>>>(y, cw, att);

  // 3) broadcast multiply (NT loads of x, NT stores of out)
  const int n4 = (B_ * C_ * HW_) / 4;
  mul_kernel<<<(n4 + 255) / 256, 256, 0, stream>>>(x, att, out, n4);
}